// MoE_51445118271909
// MI455X (gfx1250) — compile-verified
//
#include <hip/hip_runtime.h>
#include <math.h>

typedef float v2f __attribute__((ext_vector_type(2)));
typedef float v8f __attribute__((ext_vector_type(8)));
typedef float f4  __attribute__((ext_vector_type(4)));

constexpr int NB = 4096;   // tokens
constexpr int ND = 1024;   // model dim
constexpr int NH = 4096;   // hidden dim
constexpr int NE = 8;      // experts
constexpr int MT = 32;     // tokens per block tile (2 WMMA row-tiles, B reuse)
constexpr int CHUNK = 512; // H chunk per LDS pass
constexpr int XS_STRIDE = ND + 4;     // 1028 -> stride%64banks = 4 (conflict-free)
constexpr int HS_STRIDE = CHUNK + 4;  // 516  -> stride%64banks = 4

// ---- f32 WMMA 16x16x4: D = A(16x4) * B(4x16) + C(16x16) -------------------
static __device__ __forceinline__ v8f wmma4(v2f a, v2f b, v8f c) {
  // 8 args: (neg_a, A, neg_b, B, c_mod, C, reuse_a, reuse_b)
  return __builtin_amdgcn_wmma_f32_16x16x4_f32(false, a, false, b, (short)0, c,
                                               false, false);
}

// ---- kernel 1: zero output accumulator + ws counters -----------------------
__global__ void moe_init_k(float* __restrict__ out, int* __restrict__ wsi, int n) {
  int i = blockIdx.x * blockDim.x + threadIdx.x;
  if (i < n) out[i] = 0.0f;
  if (i < 16) wsi[i] = 0;  // expert counters (8 ints) + pad
}

// ---- kernel 2: top-2 gating, one wave32 per token --------------------------
__global__ __launch_bounds__(256) void moe_gate_k(
    const float* __restrict__ x, const float* __restrict__ wg,
    int* __restrict__ cnt, int* __restrict__ etok, float* __restrict__ egate,
    float* __restrict__ tgate, int* __restrict__ tidx) {
  int wv = threadIdx.x >> 5, lane = threadIdx.x & 31;
  int b = blockIdx.x * 8 + wv;
  if (b >= NB) return;

  float acc[8] = {0.f, 0.f, 0.f, 0.f, 0.f, 0.f, 0.f, 0.f};
  const float* xr = x + (size_t)b * ND;
  for (int d = lane; d < ND; d += 32) {
    float xv = xr[d];
    const f4* wr = (const f4*)(wg + (size_t)d * NE);  // 32B-aligned rows of 8
    f4 w0 = wr[0], w1 = wr[1];
    acc[0] += xv * w0.x; acc[1] += xv * w0.y; acc[2] += xv * w0.z; acc[3] += xv * w0.w;
    acc[4] += xv * w1.x; acc[5] += xv * w1.y; acc[6] += xv * w1.z; acc[7] += xv * w1.w;
  }
#pragma unroll
  for (int off = 16; off; off >>= 1) {
#pragma unroll
    for (int e = 0; e < 8; ++e) acc[e] += __shfl_xor(acc[e], off, 32);
  }
  if (lane == 0) {
    int i0 = 0; float v0 = acc[0];
#pragma unroll
    for (int e = 1; e < 8; ++e) if (acc[e] > v0) { v0 = acc[e]; i0 = e; }
    int i1 = -1; float v1 = -3.402823466e38f;
#pragma unroll
    for (int e = 0; e < 8; ++e) if (e != i0 && acc[e] > v1) { v1 = acc[e]; i1 = e; }
    float t = expf(v1 - v0);            // <= 1
    float g0 = 1.0f / (1.0f + t);
    float g1 = t / (1.0f + t);
    // per-token record (deterministic address) for the loss reduction
    tgate[b * 2 + 0] = g0; tgate[b * 2 + 1] = g1;
    tidx[b * 2 + 0]  = i0; tidx[b * 2 + 1]  = i1;
    // per-expert routing lists (order irrelevant for final output)
    int p0 = atomicAdd(&cnt[i0], 1);
    etok[i0 * NB + p0] = b; egate[i0 * NB + p0] = g0;
    int p1 = atomicAdd(&cnt[i1], 1);
    etok[i1 * NB + p1] = b; egate[i1 * NB + p1] = g1;
  }
}

// ---- kernel 3: load-balancing loss (fixed-order, deterministic) ------------
__global__ void moe_loss_k(const float* __restrict__ tgate, const int* __restrict__ tidx,
                           const float* __restrict__ coef, float* __restrict__ out) {
  int lane = threadIdx.x;  // one wave32
  float impL[8] = {0.f, 0.f, 0.f, 0.f, 0.f, 0.f, 0.f, 0.f};
  float ldL[8]  = {0.f, 0.f, 0.f, 0.f, 0.f, 0.f, 0.f, 0.f};
  for (int b = lane; b < NB; b += 32) {
#pragma unroll
    for (int k = 0; k < 2; ++k) {
      float g = tgate[b * 2 + k]; int e = tidx[b * 2 + k];
      impL[e] += g;
      if (g > 0.f) ldL[e] += 1.f;
    }
  }
#pragma unroll
  for (int off = 16; off; off >>= 1) {
#pragma unroll
    for (int e = 0; e < 8; ++e) {
      impL[e] += __shfl_xor(impL[e], off, 32);
      ldL[e]  += __shfl_xor(ldL[e],  off, 32);
    }
  }
  if (lane == 0) {
    float mi = 0.f, ml = 0.f;
#pragma unroll
    for (int e = 0; e < 8; ++e) { mi += impL[e]; ml += ldL[e]; }
    mi *= (1.0f / NE); ml *= (1.0f / NE);
    float vi = 0.f, vl = 0.f;
#pragma unroll
    for (int e = 0; e < 8; ++e) {
      float di = impL[e] - mi, dl = ldL[e] - ml;
      vi += di * di; vl += dl * dl;
    }
    vi *= (1.0f / (NE - 1)); vl *= (1.0f / (NE - 1));  // unbiased, ddof=1
    float cv = vi / (mi * mi + 1e-10f) + vl / (ml * ml + 1e-10f);
    out[(size_t)NB * ND] = cv * coef[0];
  }
}

// ---- kernel 4: fused per-expert FFN with f32 WMMA --------------------------
// grid: NE * 128 blocks (expert-major for L2 weight reuse), 256 threads.
// Block = 32-token tile; each B fragment feeds TWO wmma (rows 0-15 / 16-31),
// doubling FLOP per L2 byte (8 -> 16) and halving vmem instr per wmma.
// LDS: X[32,1024] + Hchunk[32,512] (~193 KB of the 320 KB WGP LDS).
__global__ __launch_bounds__(256) void moe_ffn_k(
    const float* __restrict__ x,
    const float* __restrict__ W1, const float* __restrict__ b1,
    const float* __restrict__ W2, const float* __restrict__ b2,
    const int* __restrict__ cnt, const int* __restrict__ etok,
    const float* __restrict__ egate, float* __restrict__ out) {
  int e = blockIdx.x >> 7;
  int tile = blockIdx.x & 127;
  int cntE = cnt[e];
  if (tile * MT >= cntE) return;  // uniform early-exit (before any barrier)

  extern __shared__ float smem[];
  float* Xs  = smem;                      // MT x XS_STRIDE
  float* Hs  = smem + MT * XS_STRIDE;     // MT x HS_STRIDE
  int*   toks = (int*)(Hs + MT * HS_STRIDE);
  float* gts  = (float*)(toks + MT);

  int tid = threadIdx.x;
  if (tid < MT) {
    int idx = tile * MT + tid;
    bool valid = idx < cntE;
    toks[tid] = etok[e * NB + (valid ? idx : tile * MT)];  // pad with row 0's token
    gts[tid]  = valid ? egate[e * NB + idx] : 0.0f;
  }
  __syncthreads();

  // gather MT token rows of x into LDS (float4, conflict-free padded stride)
  for (int idx = tid; idx < MT * (ND / 4); idx += 256) {
    int r = idx >> 8;
    int c4 = (idx & 255) << 2;
    f4 v = *(const f4*)(x + (size_t)toks[r] * ND + c4);
    *(f4*)(Xs + r * XS_STRIDE + c4) = v;
  }

  int wv = tid >> 5, lane = tid & 31;
  int half = lane >> 4, l16 = lane & 15;
  const float* W1e = W1 + (size_t)e * ND * NH;
  const float* W2e = W2 + (size_t)e * NH * ND;

  v8f acc2A[8], acc2B[8];
#pragma unroll
  for (int n = 0; n < 8; ++n) {
    acc2A[n] = (v8f){0.f, 0.f, 0.f, 0.f, 0.f, 0.f, 0.f, 0.f};
    acc2B[n] = (v8f){0.f, 0.f, 0.f, 0.f, 0.f, 0.f, 0.f, 0.f};
  }

  __syncthreads();

  for (int c = 0; c < NH / CHUNK; ++c) {
    // ---- phase 1: Hs[0:32, 0:512] = relu(X @ W1[:, c*512 : +512] + b1)
    for (int n = 0; n < 4; ++n) {
      int lcol = wv * 64 + n * 16 + l16;        // column within chunk
      int gcol = c * CHUNK + lcol;              // column within H
      const float* w1c = W1e + gcol;            // stride NH over k
      v8f accA = (v8f){0.f, 0.f, 0.f, 0.f, 0.f, 0.f, 0.f, 0.f};
      v8f accB = accA;
#pragma unroll 4
      for (int k = 0; k < ND; k += 4) {
        int kk = k + 2 * half;
        v2f a0 = *(const v2f*)(Xs + l16 * XS_STRIDE + kk);         // rows 0-15
        v2f a1 = *(const v2f*)(Xs + (16 + l16) * XS_STRIDE + kk);  // rows 16-31
        v2f bf;
        bf.x = w1c[(size_t)kk * NH];                               // B: K=kk
        bf.y = w1c[(size_t)(kk + 1) * NH];                         // B: K=kk+1
        accA = wmma4(a0, bf, accA);
        accB = wmma4(a1, bf, accB);
      }
      float bias = b1[e * NH + gcol];
#pragma unroll
      for (int i = 0; i < 8; ++i) {
        int row = i + 8 * half;                                    // C/D layout
        float vA = accA[i] + bias;
        Hs[row * HS_STRIDE + lcol] = vA > 0.f ? vA : 0.f;
        float vB = accB[i] + bias;
        Hs[(16 + row) * HS_STRIDE + lcol] = vB > 0.f ? vB : 0.f;
      }
    }
    __syncthreads();

    // ---- phase 2: acc2 += Hs @ W2[c*512 : +512, :]  (wave owns 128 d-cols)
    for (int n = 0; n < 8; ++n) {
      int dcol = wv * 128 + n * 16 + l16;
      const float* w2c = W2e + (size_t)c * CHUNK * ND + dcol;
      v8f aA = acc2A[n];
      v8f aB = acc2B[n];
#pragma unroll 4
      for (int k = 0; k < CHUNK; k += 4) {
        int kk = k + 2 * half;
        v2f a0 = *(const v2f*)(Hs + l16 * HS_STRIDE + kk);
        v2f a1 = *(const v2f*)(Hs + (16 + l16) * HS_STRIDE + kk);
        v2f bf;
        bf.x = w2c[(size_t)kk * ND];
        bf.y = w2c[(size_t)(kk + 1) * ND];
        aA = wmma4(a0, bf, aA);
        aB = wmma4(a1, bf, aB);
      }
      acc2A[n] = aA;
      acc2B[n] = aB;
    }
    __syncthreads();  // protect Hs before next chunk's phase 1
  }

  // ---- epilogue: out[tok, d] += gate * exp(o + b2)  (2 commutative adds/elt)
  int base = tile * MT;
  for (int n = 0; n < 8; ++n) {
    int dcol = wv * 128 + n * 16 + l16;
    float bias2 = b2[e * ND + dcol];
#pragma unroll
    for (int i = 0; i < 8; ++i) {
      int row = i + 8 * half;
      if (base + row < cntE) {
        float val = gts[row] * expf(acc2A[n][i] + bias2);
        atomicAdd(out + (size_t)toks[row] * ND + dcol, val);
      }
      int row2 = row + 16;
      if (base + row2 < cntE) {
        float val = gts[row2] * expf(acc2B[n][i] + bias2);
        atomicAdd(out + (size_t)toks[row2] * ND + dcol, val);
      }
    }
  }
}

// ---- kernel 5: y = log(c == 0 ? EPS : c) -----------------------------------
__global__ void moe_fin_k(float* __restrict__ out, int n) {
  int i = blockIdx.x * blockDim.x + threadIdx.x;
  if (i < n) {
    float c = out[i];
    out[i] = logf(c == 0.0f ? 2.220446049250313e-16f : c);
  }
}

extern "C" void kernel_launch(void* const* d_in, const int* in_sizes, int n_in,
                              void* d_out, int out_size, void* d_ws, size_t ws_size,
                              hipStream_t stream) {
  const float* x    = (const float*)d_in[0];
  const float* wg   = (const float*)d_in[1];
  const float* W1   = (const float*)d_in[2];
  const float* b1   = (const float*)d_in[3];
  const float* W2   = (const float*)d_in[4];
  const float* b2   = (const float*)d_in[5];
  const float* coef = (const float*)d_in[6];
  float* out = (float*)d_out;

  // workspace layout
  char* ws = (char*)d_ws;
  int*   cnt   = (int*)ws;                                 // 8 ints (+pad to 64B)
  int*   etok  = (int*)(ws + 64);                          // NE*NB ints
  float* egate = (float*)(ws + 64 + sizeof(int) * NE * NB);
  float* tgate = (float*)(ws + 64 + (sizeof(int) + sizeof(float)) * NE * NB);
  int*   tidx  = (int*)((char*)tgate + sizeof(float) * NB * 2);

  const int nOut = NB * ND;

  moe_init_k<<<(nOut + 1 + 255) / 256, 256, 0, stream>>>(out, (int*)ws, nOut + 1);
  moe_gate_k<<<NB / 8, 256, 0, stream>>>(x, wg, cnt, etok, egate, tgate, tidx);
  moe_loss_k<<<1, 32, 0, stream>>>(tgate, tidx, coef, out);

  size_t smem = (size_t)(MT * XS_STRIDE + MT * HS_STRIDE) * sizeof(float)
              + MT * sizeof(int) + MT * sizeof(float);
  (void)hipFuncSetAttribute((const void*)moe_ffn_k,
                            hipFuncAttributeMaxDynamicSharedMemorySize, (int)smem);
  moe_ffn_k<<<NE * (NB / MT), 256, smem, stream>>>(x, W1, b1, W2, b2, cnt, etok,
                                                   egate, out);

  moe_fin_k<<<(nOut + 255) / 256, 256, 0, stream>>>(out, nOut);
}